// InterpolateNd_77412490543530
// MI455X (gfx1250) — compile-verified
//
#include <hip/hip_runtime.h>
#include <hip/hip_bf16.h>

// ---------------------------------------------------------------------------
// 4-D linear upsample x2 (align_corners=False), fused single pass.
//   In : [8][32][32][32][32] f32  (33.5 MB, L2-resident -> 16x reuse is free)
//   Out: [8][64][64][64][64] f32  (537 MB, streamed once, NT stores)
// Roofline: ~570 MB @ 23.3 TB/s  ->  ~24.5 us. Bandwidth-bound; WMMA would
// waste 16x MACs (interp matrix is 2-sparse) and force extra passes.
// CDNA5 paths used: global_load_async_to_lds_b128 (ASYNCcnt), s_wait_asynccnt,
// wave32-sized mapping, non-temporal b128 stores.
// ---------------------------------------------------------------------------

typedef float v4f __attribute__((ext_vector_type(4)));
typedef int   v4i __attribute__((vector_size(16)));   // matches builtin param type

#define ROWP 36                     // padded LDS row stride (floats): 144 B, 16B-aligned, bank-spread
#define TILE_FLOATS (9 * 32 * ROWP) // 41,472 B LDS per workgroup

__device__ __forceinline__ void async_ld16(const float* gsrc, float* ldst) {
#if defined(__has_builtin) && __has_builtin(__builtin_amdgcn_global_load_async_to_lds_b128)
    // Signature (from hipcc diagnostic): param0 = AS1 v4i* (source, global),
    // then AS3 v4i* (dest, LDS), imm offset, imm cpol.
    __builtin_amdgcn_global_load_async_to_lds_b128(
        (__attribute__((address_space(1))) v4i*)gsrc,
        (__attribute__((address_space(3))) v4i*)ldst,
        0, 0);
#else
    // ISA 15.18.3 opcode 98: vdst = LDS byte address, vaddr = 64-bit global addr
    asm volatile("global_load_async_to_lds_b128 %0, %1, off"
                 :: "v"((unsigned)(unsigned long long)ldst),
                    "v"((unsigned long long)gsrc)
                 : "memory");
#endif
}

__device__ __forceinline__ void wait_async0() {
#if defined(__has_builtin) && __has_builtin(__builtin_amdgcn_s_wait_asynccnt)
    __builtin_amdgcn_s_wait_asynccnt(0);
#else
    asm volatile("s_wait_asynccnt 0x0" ::: "memory");
#endif
}

extern "C" __global__ __launch_bounds__(256)
void interp4d_x2_kernel(const float* __restrict__ in, float* __restrict__ out)
{
    __shared__ __align__(16) float tile[TILE_FLOATS]; // [9 dd][32 i3][ROWP]

    const int wg = blockIdx.x;           // 8 * 32 * 32 = 8192 workgroups
    const int c  = wg >> 10;             // channel 0..7
    const int a1 = (wg >> 5) & 31;       // out x1 pair index
    const int a2 = wg & 31;              // out x2 pair index
    const int t  = threadIdx.x;          // 256 threads = 8 waves (wave32)

    // Clamped source slots along axes 1/2 (duplicate-at-edge makes generic
    // 0.25/0.75 weights reproduce the reference's clamped-src semantics).
    const int i1s[3] = { (a1 > 0) ? a1 - 1 : 0, a1, (a1 < 31) ? a1 + 1 : 31 };
    const int i2s[3] = { (a2 > 0) ? a2 - 1 : 0, a2, (a2 < 31) ? a2 + 1 : 31 };

    // ---- Stage 3x3x32x32 input neighborhood into LDS via async copies ----
    {
        const int q4 = (t & 7) * 4;      // float offset within a 32-float row
        const int r3 = t >> 3;           // 0..31
        #pragma unroll
        for (int k = 0; k < 9; ++k) {    // dd = (d1,d2) slot, compile-time
            const int d1 = k / 3, d2 = k % 3;
            const float* src = in + ((((c * 32 + i1s[d1]) * 32 + i2s[d2]) * 32 + r3) * 32 + q4);
            float*       dst = tile + ((k * 32 + r3) * ROWP + q4);
            async_ld16(src, dst);
        }
    }
    wait_async0();
    __syncthreads();

    // ---- Per-thread output row: (ox1, ox2, x3out) fixed, x4 = 0..63 ----
    const int ox1 = t >> 7;
    const int ox2 = (t >> 6) & 1;
    const int x3o = t & 63;
    const int a3  = x3o >> 1, r3o = x3o & 1;

    const float w1h = ox1 ? 0.25f : 0.75f, w1l = 1.0f - w1h;
    const float w2h = ox2 ? 0.25f : 0.75f, w2l = 1.0f - w2h;
    const float w3h = r3o ? 0.25f : 0.75f, w3l = 1.0f - w3h;
    const int   i3lo = (a3 + r3o - 1 < 0) ? 0 : a3 + r3o - 1;
    const int   i3hi = (a3 + r3o > 31) ? 31 : a3 + r3o;

    float coef[8];
    int   rbase[8];
    #pragma unroll
    for (int e = 0; e < 8; ++e) {
        const int e1 = e >> 2, e2 = (e >> 1) & 1, e3 = e & 1;
        const int dd = (ox1 + e1) * 3 + (ox2 + e2);
        const int i3 = e3 ? i3hi : i3lo;
        rbase[e] = (dd * 32 + i3) * ROWP;
        coef[e]  = (e1 ? w1h : w1l) * (e2 ? w2h : w2l) * (e3 ? w3h : w3l);
    }

    // Weighted combine of 8 LDS rows -> 32-float register row (8 FMA/elem).
    float a[32];
    #pragma unroll
    for (int m = 0; m < 8; ++m) {
        v4f s = 0.0f;
        #pragma unroll
        for (int e = 0; e < 8; ++e) {
            const v4f v = *(const v4f*)(tile + rbase[e] + 4 * m); // ds_load_b128
            s += coef[e] * v;
        }
        a[4 * m + 0] = s.x; a[4 * m + 1] = s.y;
        a[4 * m + 2] = s.z; a[4 * m + 3] = s.w;
    }

    // Expand 32 -> 64 along x4 and stream out (non-temporal b128 stores).
    float* orow = out + (size_t)((((c * 64 + (2 * a1 + ox1)) * 64 + (2 * a2 + ox2)) * 64 + x3o)) * 64;
    #pragma unroll
    for (int m = 0; m < 16; ++m) {
        v4f o;
        #pragma unroll
        for (int j = 0; j < 4; ++j) {
            const int   x4 = 4 * m + j;
            const int   a4 = x4 >> 1, r = x4 & 1;
            const int   il = (a4 + r - 1 < 0) ? 0 : a4 + r - 1;
            const int   ih = (a4 + r > 31) ? 31 : a4 + r;
            const float wh = r ? 0.25f : 0.75f;
            o[j] = fmaf(a[il], 1.0f - wh, a[ih] * wh);
        }
        __builtin_nontemporal_store(o, (v4f*)(orow + 4 * m));
    }
}

extern "C" void kernel_launch(void* const* d_in, const int* in_sizes, int n_in,
                              void* d_out, int out_size, void* d_ws, size_t ws_size,
                              hipStream_t stream) {
    (void)in_sizes; (void)n_in; (void)out_size; (void)d_ws; (void)ws_size;
    const float* in  = (const float*)d_in[0];
    float*       o   = (float*)d_out;
    // 8 channels * 32 x1-pairs * 32 x2-pairs = 8192 workgroups of 256 threads.
    hipLaunchKernelGGL(interp4d_x2_kernel, dim3(8192), dim3(256), 0, stream, in, o);
}